// StitchingCurrents_23733989277805
// MI455X (gfx1250) — compile-verified
//
#include <hip/hip_runtime.h>

// ---------------------------------------------------------------------------
// StitchingCurrents energy on MI455X (gfx1250, wave32).
//
// E = 1.8*S(s,s) + 1.8*S(t,t) + 2*S(r,r) - 2*S(r,s) - 2*S(r,t) - 1.6*S(s,t)
// S(A,B) = sum_ij (nA_i . nB_j) / (1 + ||cA_i - cB_j||^2)^2      (SIGMA = 1)
//
// Per 16x16 pair tile (one wave):
//   ndot = A_n(16x3) @ B_n(3x16)                 -> 1x V_WMMA_F32_16X16X4_F32
//   d    = [c,|c|^2,1] @ [-2c,1,|c|^2]^T (K=5)   -> 2x V_WMMA_F32_16X16X4_F32
//   epilogue: accv += ndot * v_rcp_f32((1+d)^2)  -> packed VALU, no deps chain
//
// B operands are pre-swizzled into WMMA lane layout by the prep kernel so the
// inner loop is: 2 loads (b128+b64) + 2 prefetches + 3 WMMA + packed epilogue.
// ---------------------------------------------------------------------------

typedef __attribute__((ext_vector_type(2))) float v2f;
typedef __attribute__((ext_vector_type(8))) float v8f;

// ---------------------------------------------------------------------------
// Prep helpers
// ---------------------------------------------------------------------------
__device__ __forceinline__ void face_nc(const float* __restrict__ v,
                                        const int*   __restrict__ idx,
                                        int f, float4& cOut, float4& nOut) {
  const int i0 = idx[3 * f + 0], i1 = idx[3 * f + 1], i2 = idx[3 * f + 2];
  const float ax = v[3 * i0], ay = v[3 * i0 + 1], az = v[3 * i0 + 2];
  const float bx = v[3 * i1], by = v[3 * i1 + 1], bz = v[3 * i1 + 2];
  const float cx = v[3 * i2], cy = v[3 * i2 + 1], cz = v[3 * i2 + 2];
  const float ux = ax - bx, uy = ay - by, uz = az - bz;
  const float wx = cx - bx, wy = cy - by, wz = cz - bz;
  const float nx = 0.5f * (uy * wz - uz * wy);
  const float ny = 0.5f * (uz * wx - ux * wz);
  const float nz = 0.5f * (ux * wy - uy * wx);
  const float mx = (ax + bx + cx) * (1.0f / 3.0f);
  const float my = (ay + by + cy) * (1.0f / 3.0f);
  const float mz = (az + bz + cz) * (1.0f / 3.0f);
  cOut = make_float4(mx, my, mz, mx * mx + my * my + mz * mz);
  nOut = make_float4(nx, ny, nz, 0.0f);
}

// WMMA f32 16x16x4 B layout (mirrors A halves): lane<16 holds column N=lane
// with (K0,K1); lane>=16 holds column N=lane-16 with (K2,K3).
// Per face f we emit both the lo-lane and hi-lane operand halves:
//   lo: b_n=(n.x,n.y)  b_d0=(-2c.x,-2c.y)  b_d1=(|c|^2, 0)
//   hi: b_n=(n.z,0)    b_d0=(-2c.z, 1)     b_d1=(0, 0)
// Packed as B4 = {b_n.x, b_n.y, b_d0.x, b_d0.y}, B2 = {b_d1.x, b_d1.y}.
__device__ __forceinline__ void write_bop(float4* __restrict__ B4,
                                          float2* __restrict__ B2, int f,
                                          const float4& c, const float4& n) {
  const int jt = f >> 4, sub = f & 15;
  const int lo = jt * 32 + sub, hi = lo + 16;
  B4[lo] = make_float4(n.x, n.y, -2.0f * c.x, -2.0f * c.y);
  B2[lo] = make_float2(c.w, 0.0f);
  B4[hi] = make_float4(n.z, 0.0f, -2.0f * c.z, 1.0f);
  B2[hi] = make_float2(0.0f, 0.0f);
}

__global__ void prep_kernel(const float* __restrict__ sv, const int* __restrict__ si,
                            const float* __restrict__ tv, const int* __restrict__ ti,
                            const float* __restrict__ rn, const float* __restrict__ rc,
                            float4* __restrict__ srcC, float4* __restrict__ srcN,
                            float4* __restrict__ tarC, float4* __restrict__ tarN,
                            float4* __restrict__ refC, float4* __restrict__ refN,
                            float4* __restrict__ srcB4, float2* __restrict__ srcB2,
                            float4* __restrict__ tarB4, float2* __restrict__ tarB2,
                            float4* __restrict__ refB4, float2* __restrict__ refB2,
                            int FS, int FT, int FR) {
  const int f = blockIdx.x * blockDim.x + threadIdx.x;
  if (f < FS) {
    float4 c, n; face_nc(sv, si, f, c, n);
    srcC[f] = c; srcN[f] = n; write_bop(srcB4, srcB2, f, c, n);
  }
  if (f < FT) {
    float4 c, n; face_nc(tv, ti, f, c, n);
    tarC[f] = c; tarN[f] = n; write_bop(tarB4, tarB2, f, c, n);
  }
  if (f < FR) {
    const float x = rc[3 * f], y = rc[3 * f + 1], z = rc[3 * f + 2];
    float4 c = make_float4(x, y, z, x * x + y * y + z * z);
    float4 n = make_float4(rn[3 * f], rn[3 * f + 1], rn[3 * f + 2], 0.0f);
    refC[f] = c; refN[f] = n; write_bop(refB4, refB2, f, c, n);
  }
}

// ---------------------------------------------------------------------------
// Pair-energy kernel: one block per 16-row A tile; each of 8 waves streams
// every 8th 16-col B tile. Requires nA, nB multiples of 16 (8192 here).
// ---------------------------------------------------------------------------
__global__ __launch_bounds__(256) void pair_energy_kernel(
    const float4* __restrict__ Ac, const float4* __restrict__ An,
    const float4* __restrict__ B4, const float2* __restrict__ B2,
    int nB, float coef, float* __restrict__ partials) {
  __shared__ float wave_sums[8];
  const int  lane = threadIdx.x & 31;
  const int  wid  = threadIdx.x >> 5;
  const int  sub  = lane & 15;
  const bool hi   = lane >= 16;

  // A operands, built once per wave (f32 16x16x4 A layout: lanes 0-15 hold
  // M=lane with (K0,K1); lanes 16-31 hold M=lane-16 with (K2,K3)).
  const int    m  = blockIdx.x * 16 + sub;
  const float4 ca = Ac[m];
  const float4 na = An[m];
  v2f a_n, a_d0, a_d1;
  a_n.x  = hi ? na.z : na.x;   a_n.y  = hi ? 0.0f : na.y;   // [nx,ny,nz,0]
  a_d0.x = hi ? ca.z : ca.x;   a_d0.y = hi ? ca.w : ca.y;   // [cx,cy,cz,|c|^2]
  a_d1.x = hi ? 0.0f : 1.0f;   a_d1.y = 0.0f;               // [1,0,0,0]

  v8f accv = {};
  const int tilesB = nB >> 4;
#pragma unroll 2
  for (int jt = wid; jt < tilesB; jt += 8) {
    const int    bi = jt * 32 + lane;      // pre-swizzled B operand slot
    const float4 q  = B4[bi];
    const float2 q2 = B2[bi];
    __builtin_prefetch(&B4[bi + 256], 0, 1);  // this wave's next tile
    __builtin_prefetch(&B2[bi + 256], 0, 1);

    v2f b_n, b_d0, b_d1;
    b_n.x  = q.x;  b_n.y  = q.y;
    b_d0.x = q.z;  b_d0.y = q.w;
    b_d1.x = q2.x; b_d1.y = q2.y;

    v8f zero = {};
    v8f ndm = __builtin_amdgcn_wmma_f32_16x16x4_f32(false, a_n,  false, b_n,
                                                    (short)0, zero, false, false);
    v8f dm  = __builtin_amdgcn_wmma_f32_16x16x4_f32(false, a_d0, false, b_d0,
                                                    (short)0, zero, false, false);
    dm      = __builtin_amdgcn_wmma_f32_16x16x4_f32(false, a_d1, false, b_d1,
                                                    (short)0, dm,   false, false);

    // Cauchy weight + vector accumulate (independent packed FMAs, no chain).
    v8f w;
#pragma unroll
    for (int e = 0; e < 8; ++e) {
      const float t = 1.0f + dm[e];
      w[e] = __builtin_amdgcn_rcpf(t * t);
    }
    accv = accv + ndm * w;
  }

  // Deterministic reduction: tree over v8f, wave shuffle, LDS, block partial.
  float acc = ((accv[0] + accv[1]) + (accv[2] + accv[3])) +
              ((accv[4] + accv[5]) + (accv[6] + accv[7]));
  for (int off = 16; off > 0; off >>= 1) acc += __shfl_xor(acc, off, 32);
  if (lane == 0) wave_sums[wid] = acc;
  __syncthreads();
  if (threadIdx.x == 0) {
    float s = 0.0f;
    for (int w8 = 0; w8 < 8; ++w8) s += wave_sums[w8];
    partials[blockIdx.x] = coef * s;
  }
}

__global__ void zero_kernel(float* __restrict__ p, int n) {
  const int i = blockIdx.x * blockDim.x + threadIdx.x;
  if (i < n) p[i] = 0.0f;
}

__global__ __launch_bounds__(256) void final_reduce_kernel(
    const float* __restrict__ partials, int n, float* __restrict__ out) {
  __shared__ float sh[256];
  float s = 0.0f;
  for (int i = threadIdx.x; i < n; i += 256) s += partials[i];
  sh[threadIdx.x] = s;
  __syncthreads();
  for (int off = 128; off > 0; off >>= 1) {
    if (threadIdx.x < off) sh[threadIdx.x] += sh[threadIdx.x + off];
    __syncthreads();
  }
  if (threadIdx.x == 0) out[0] = sh[0];
}

// ---------------------------------------------------------------------------
extern "C" void kernel_launch(void* const* d_in, const int* in_sizes, int n_in,
                              void* d_out, int out_size, void* d_ws, size_t ws_size,
                              hipStream_t stream) {
  const float* sv = (const float*)d_in[0];
  const int*   si = (const int*)  d_in[1];
  const float* tv = (const float*)d_in[2];
  const int*   ti = (const int*)  d_in[3];
  const float* rn = (const float*)d_in[4];
  const float* rc = (const float*)d_in[5];
  float* out = (float*)d_out;

  const int FS = in_sizes[1] / 3;
  const int FT = in_sizes[3] / 3;
  const int FR = in_sizes[4] / 3;

  // Workspace layout: all float4 arrays first (16B aligned), then float2,
  // then the per-block partial sums. Total ~1.9 MB for 3x8192 faces.
  float4* ws4 = (float4*)d_ws;
  size_t o = 0;
  float4* srcC  = ws4 + o; o += (size_t)FS;
  float4* srcN  = ws4 + o; o += (size_t)FS;
  float4* tarC  = ws4 + o; o += (size_t)FT;
  float4* tarN  = ws4 + o; o += (size_t)FT;
  float4* refC  = ws4 + o; o += (size_t)FR;
  float4* refN  = ws4 + o; o += (size_t)FR;
  float4* srcB4 = ws4 + o; o += (size_t)(2 * FS);
  float4* tarB4 = ws4 + o; o += (size_t)(2 * FT);
  float4* refB4 = ws4 + o; o += (size_t)(2 * FR);
  float2* f2    = (float2*)(ws4 + o);
  size_t o2 = 0;
  float2* srcB2 = f2 + o2; o2 += (size_t)(2 * FS);
  float2* tarB2 = f2 + o2; o2 += (size_t)(2 * FT);
  float2* refB2 = f2 + o2; o2 += (size_t)(2 * FR);
  float* partials = (float*)(f2 + o2);

  zero_kernel<<<(out_size + 63) / 64, 64, 0, stream>>>(out, out_size);

  int maxF = FS > FT ? FS : FT; if (FR > maxF) maxF = FR;
  prep_kernel<<<(maxF + 255) / 256, 256, 0, stream>>>(
      sv, si, tv, ti, rn, rc, srcC, srcN, tarC, tarN, refC, refN,
      srcB4, srcB2, tarB4, tarB2, refB4, refB2, FS, FT, FR);

  const int gS = FS / 16, gT = FT / 16, gR = FR / 16;
  int po = 0;
  pair_energy_kernel<<<gS, 256, 0, stream>>>(srcC, srcN, srcB4, srcB2, FS,  1.8f, partials + po); po += gS;
  pair_energy_kernel<<<gT, 256, 0, stream>>>(tarC, tarN, tarB4, tarB2, FT,  1.8f, partials + po); po += gT;
  pair_energy_kernel<<<gR, 256, 0, stream>>>(refC, refN, refB4, refB2, FR,  2.0f, partials + po); po += gR;
  pair_energy_kernel<<<gR, 256, 0, stream>>>(refC, refN, srcB4, srcB2, FS, -2.0f, partials + po); po += gR;
  pair_energy_kernel<<<gR, 256, 0, stream>>>(refC, refN, tarB4, tarB2, FT, -2.0f, partials + po); po += gR;
  pair_energy_kernel<<<gS, 256, 0, stream>>>(srcC, srcN, tarB4, tarB2, FT, -1.6f, partials + po); po += gS;

  final_reduce_kernel<<<1, 256, 0, stream>>>(partials, po, out);
}